// ChebConvGAD_Hetero_36043365548320
// MI455X (gfx1250) — compile-verified
//
#include <hip/hip_runtime.h>

typedef __attribute__((ext_vector_type(2))) float v2f;
typedef __attribute__((ext_vector_type(8))) float v8f;

#define NODES 100000
#define EDGES 1000000
#define RELS  3
#define HF    64

// ---------------------------------------------------------------------------
// Degree count per relation: deg[r][dst] += 1  (float atomics; exact for deg<2^24)
// ---------------------------------------------------------------------------
__global__ void k_deg(const int* __restrict__ dst, float* __restrict__ deg) {
  int tid = blockIdx.x * blockDim.x + threadIdx.x;
  if (tid >= RELS * EDGES) return;
  int r = tid / EDGES;
  unsafeAtomicAdd(&deg[(size_t)r * NODES + dst[tid]], 1.0f);
}

// In-place: deg -> clamp(deg,1)^-0.5
__global__ void k_dinv(float* __restrict__ deg) {
  int i = blockIdx.x * blockDim.x + threadIdx.x;
  if (i >= RELS * NODES) return;
  float v = deg[i];
  v = v < 1.0f ? 1.0f : v;
  deg[i] = 1.0f / sqrtf(v);
}

// ---------------------------------------------------------------------------
// Gather + segment-sum: ah[dst] += feat[src] * dinv[src]
// 16 threads per edge, 4 features each (float4 gather, 4 f32 atomics).
// Node arrays (25.6 MB) are L2-resident on MI455X (192 MB L2).
// ---------------------------------------------------------------------------
__global__ void k_scatter(const float* __restrict__ feat,
                          const float* __restrict__ dinv,
                          const int* __restrict__ src,
                          const int* __restrict__ dst,
                          float* __restrict__ ah) {
  int tid = blockIdx.x * blockDim.x + threadIdx.x;
  if (tid >= EDGES * 16) return;
  int e = tid >> 4;
  int c = (tid & 15) << 2;
  int s = src[e];
  int d = dst[e];
  float dv = dinv[s];
  const float4 f = *(const float4*)(feat + (size_t)s * HF + c);
  float* o = ah + (size_t)d * HF + c;
  unsafeAtomicAdd(o + 0, f.x * dv);
  unsafeAtomicAdd(o + 1, f.y * dv);
  unsafeAtomicAdd(o + 2, f.z * dv);
  unsafeAtomicAdd(o + 3, f.w * dv);
}

// ---------------------------------------------------------------------------
// Fused node-feature GEMM using V_WMMA_F32_16X16X4_F32 (full fp32 precision).
//   out = A1 @ W[0:64]  (+ (dinv*row_sign) scaled A2 @ W[64:128])  + bias
//   optional leaky-ReLU; optional accum += out (each row owned by one wave).
//
// W is staged in LDS pre-swizzled as K-pairs:
//   sW[((k>>1)*64 + col)*2 + (k&1)] = W[k][col]
// so a B fragment (two consecutive K rows at one column) is a single aligned
// ds_load_b64 into an even VGPR pair -> no register-packing moves before WMMA.
//
// Fragment layouts per CDNA5 ISA 7.12.2:
//   A (16x4 f32): lane L holds A[L%16][k0 + 2*(L>>4) + j] in vgpr j
//   B (4x16 f32): lane L holds B[k0 + 2*(L>>4) + j][L%16] in vgpr j
//   C (16x16 f32): vgpr v, lane L -> C[v + 8*(L>>4)][L%16]
// ---------------------------------------------------------------------------
__launch_bounds__(256)
__global__ void k_gemm(const float* __restrict__ A1,
                       const float* __restrict__ A2,     // nullable
                       const float* __restrict__ dinv,   // nullable per-row scale for A2
                       float a2sign,
                       const float* __restrict__ W,      // (dual?128:64) x 64, K-major
                       const float* __restrict__ bias,   // 64
                       float* __restrict__ out,
                       float* __restrict__ accum,        // nullable: accum += result
                       int n, int act) {
  __shared__ float sW[128 * 64];
  __shared__ float sB[64];
  const bool dual = (A2 != nullptr);
  const int wtot = (dual ? 128 : 64) * 64;
  // K-pair swizzle: element W[k][col] -> sW[(k>>1)*128 + col*2 + (k&1)]
  for (int i = threadIdx.x; i < wtot; i += 256) {
    int k = i >> 6;
    int col = i & 63;
    sW[(k >> 1) * 128 + col * 2 + (k & 1)] = W[i];
  }
  if (threadIdx.x < 64) sB[threadIdx.x] = bias[threadIdx.x];
  __syncthreads();

  const int lane = threadIdx.x & 31;
  const int wave = threadIdx.x >> 5;
  const int strip = blockIdx.x * 8 + wave;      // N % 16 == 0 -> whole strips only
  if (strip >= (n >> 4)) return;                // wave-uniform; EXEC stays all-ones
  const int row0 = strip << 4;
  const int m  = lane & 15;                     // A row in tile / B,C column
  const int kh = (lane >> 4) << 1;              // K sub-offset within 4-group: 0 or 2

  const float* a1p = A1 + (size_t)(row0 + m) * HF + kh;
  v8f acc[4] = {};

#pragma unroll
  for (int k0 = 0; k0 < 64; k0 += 4) {
    v2f a = *(const v2f*)(a1p + k0);
    // swizzled base for K rows (k0+kh, k0+kh+1):
    const float* wb = &sW[(k0 + kh) * 64 + m * 2];
#pragma unroll
    for (int t = 0; t < 4; ++t) {
      v2f b = *(const v2f*)(wb + t * 32);
      acc[t] = __builtin_amdgcn_wmma_f32_16x16x4_f32(
          false, a, false, b, (short)0, acc[t], false, false);
    }
  }
  if (dual) {
    const float* a2p = A2 + (size_t)(row0 + m) * HF + kh;
    const float ds = (dinv ? dinv[row0 + m] : 1.0f) * a2sign;
#pragma unroll
    for (int k0 = 0; k0 < 64; k0 += 4) {
      v2f a = *(const v2f*)(a2p + k0);
      a = a * ds;
      const float* wb = &sW[(64 + k0 + kh) * 64 + m * 2];
#pragma unroll
      for (int t = 0; t < 4; ++t) {
        v2f b = *(const v2f*)(wb + t * 32);
        acc[t] = __builtin_amdgcn_wmma_f32_16x16x4_f32(
            false, a, false, b, (short)0, acc[t], false, false);
      }
    }
  }

  const int rbase = row0 + ((lane >> 4) << 3);  // +8 rows for upper lane half
#pragma unroll
  for (int t = 0; t < 4; ++t) {
    const float bv = sB[t * 16 + m];
#pragma unroll
    for (int v = 0; v < 8; ++v) {
      float val = acc[t][v] + bv;
      if (act) val = val > 0.0f ? val : 0.01f * val;
      size_t o = (size_t)(rbase + v) * HF + t * 16 + m;
      out[o] = val;
      if (accum) accum[o] += val;   // row owned exclusively by this wave
    }
  }
}

// ---------------------------------------------------------------------------
// Head: out[N,2] = leaky_relu(h_all) @ W4 + b4
// ---------------------------------------------------------------------------
__global__ void k_head(const float* __restrict__ hall,
                       const float* __restrict__ W4,
                       const float* __restrict__ b4,
                       float* __restrict__ out, int n) {
  __shared__ float sW[128];
  __shared__ float sb[2];
  if (threadIdx.x < 128) sW[threadIdx.x] = W4[threadIdx.x];
  if (threadIdx.x < 2) sb[threadIdx.x] = b4[threadIdx.x];
  __syncthreads();
  int i = blockIdx.x * blockDim.x + threadIdx.x;
  if (i >= n) return;
  const float* hp = hall + (size_t)i * HF;
  float a0 = sb[0], a1 = sb[1];
#pragma unroll
  for (int k = 0; k < 64; ++k) {
    float v = hp[k];
    v = v > 0.0f ? v : 0.01f * v;
    a0 += v * sW[2 * k];
    a1 += v * sW[2 * k + 1];
  }
  out[2 * (size_t)i + 0] = a0;
  out[2 * (size_t)i + 1] = a1;
}

// ---------------------------------------------------------------------------
extern "C" void kernel_launch(void* const* d_in, const int* in_sizes, int n_in,
                              void* d_out, int out_size, void* d_ws, size_t ws_size,
                              hipStream_t stream) {
  const float* x   = (const float*)d_in[0];
  const int*   src = (const int*)d_in[1];
  const int*   dst = (const int*)d_in[2];
  const float* W1  = (const float*)d_in[3];
  const float* b1  = (const float*)d_in[4];
  const float* W2  = (const float*)d_in[5];
  const float* b2  = (const float*)d_in[6];
  const float* Wc1 = (const float*)d_in[7];
  const float* bc1 = (const float*)d_in[8];
  const float* Wc2 = (const float*)d_in[9];
  const float* bc2 = (const float*)d_in[10];
  const float* W3  = (const float*)d_in[11];
  const float* b3  = (const float*)d_in[12];
  const float* W4  = (const float*)d_in[13];
  const float* b4  = (const float*)d_in[14];
  float* out = (float*)d_out;

  const size_t NH = (size_t)NODES * HF;
  float* ws   = (float*)d_ws;
  float* hbuf = ws;            // N*H  current node features
  float* h0   = hbuf + NH;     // N*H
  float* h1   = h0 + NH;       // N*H
  float* hall = h1 + NH;       // N*H
  float* ah   = hall + NH;     // N*H  segment-sum accumulator
  float* dinv = ah + NH;       // R*N  degrees -> dinv (in place)

  // Degrees / normalization (depends only on dst, computed once per launch).
  hipMemsetAsync(dinv, 0, sizeof(float) * (size_t)RELS * NODES, stream);
  k_deg<<<(RELS * EDGES + 255) / 256, 256, 0, stream>>>(dst, dinv);
  k_dinv<<<(RELS * NODES + 255) / 256, 256, 0, stream>>>(dinv);

  const int gblocks = ((NODES / 16) + 7) / 8;  // 8 waves (16-row strips) per block

  // h = lrelu(lrelu(x@W1+b1)@W2+b2)
  k_gemm<<<gblocks, 256, 0, stream>>>(x,  nullptr, nullptr, 1.0f, W1, b1, h0,   nullptr, NODES, 1);
  k_gemm<<<gblocks, 256, 0, stream>>>(h0, nullptr, nullptr, 1.0f, W2, b2, hbuf, nullptr, NODES, 1);
  hipMemsetAsync(hall, 0, sizeof(float) * NH, stream);

  for (int r = 0; r < RELS; ++r) {
    const int*   sr = src + (size_t)r * EDGES;
    const int*   dr = dst + (size_t)r * EDGES;
    const float* dv = dinv + (size_t)r * NODES;

    // h0 = [h, -D^-1/2 A D^-1/2 h] @ Wc1 + bc1
    hipMemsetAsync(ah, 0, sizeof(float) * NH, stream);
    k_scatter<<<(EDGES * 16 + 255) / 256, 256, 0, stream>>>(hbuf, dv, sr, dr, ah);
    k_gemm<<<gblocks, 256, 0, stream>>>(hbuf, ah, dv, -1.0f, Wc1, bc1, h0, nullptr, NODES, 0);

    // h1 = [h0, -D^-1/2 A D^-1/2 h0] @ Wc2 + bc2
    hipMemsetAsync(ah, 0, sizeof(float) * NH, stream);
    k_scatter<<<(EDGES * 16 + 255) / 256, 256, 0, stream>>>(h0, dv, sr, dr, ah);
    k_gemm<<<gblocks, 256, 0, stream>>>(h0, ah, dv, -1.0f, Wc2, bc2, h1, nullptr, NODES, 0);

    // h = [h0, h1] @ W3 + b3 ;  h_all += h
    k_gemm<<<gblocks, 256, 0, stream>>>(h0, h1, nullptr, 1.0f, W3, b3, hbuf, hall, NODES, 0);
  }

  k_head<<<(NODES + 255) / 256, 256, 0, stream>>>(hall, W4, b4, out, NODES);
}